// ApproxEMDLoss_7490422964952
// MI455X (gfx1250) — compile-verified
//
#include <hip/hip_runtime.h>

// ApproxEMD loss on MI455X (gfx1250, wave32) — multi-kernel, high-occupancy version.
//
// Per level (kernel boundaries provide the per-batch sync the algorithm needs):
//   A   : flash-softmax (online max/sum) + column sums of w and w*d2   [256 WGs]
//   Mid : fac = min(satr/(colsum+1e-9),1); satr update; loss += fac*colsum(w*d2)
//   C   : row sums of w*fac -> satl update                              [256 WGs]
//   Post: lsr = log2(satr+1e-30); zero column accumulators
//
// d2 tiles via V_WMMA_F32_16X16X4_F32 with C=0:
//   A row = [-2x0,-2x1,-2x2,|x|^2], B col = [y0,y1,y2,1]  =>  dp = |x|^2 - 2x.y
//   d2 = dp + |y|^2 folded algebraically (no C-fragment broadcast).
// All exps in base-2 domain via native v_exp_f32 / v_log_f32:
//   level2 = level*log2(e), lsr stores log2(satr+1e-30).

typedef __attribute__((ext_vector_type(2))) float v2f;
typedef __attribute__((ext_vector_type(8))) float v8f;

#define NB      16
#define NPTS    2048
#define MPTS    2048
#define CTILES  (MPTS / 16)
#define LEVELS  44
#define WGPB    16            // workgroups per batch (row split)
#define TPB     256           // 8 waves
#define ROWS_PER_WG (NPTS / WGPB)   // 128 rows -> 8 waves x 16-row blocks

#define EXP2(x) __builtin_amdgcn_exp2f(x)   // v_exp_f32 (2^x)
#define LOG2(x) __builtin_amdgcn_logf(x)    // v_log_f32 (log2 x)

// Persistent state in device globals (~2.2 MB, L2-resident).
__device__ float g_xs4[NB * NPTS * 4];   // (-2x0,-2x1,-2x2,|x|^2)
__device__ float g_ys4[NB * MPTS * 4];   // (y0,y1,y2,1)
__device__ float g_ynorm[NB * MPTS];     // |y|^2
__device__ float g_satl[NB * NPTS];
__device__ float g_satr[NB * MPTS];
__device__ float g_lsr[NB * MPTS];       // log2(satr + 1e-30)
__device__ float g_ssum[NB * MPTS];      // colsum of w (per level)
__device__ float g_wd2[NB * MPTS];       // colsum of w*d2 (per level)
__device__ float g_sfac[NB * MPTS];     // column rescale factor
__device__ float g_rowM[NB * NPTS];      // row max (base-2 logits)
__device__ float g_winv[NB * NPTS];      // satl[row] / row exp-sum

__global__ __launch_bounds__(TPB) void emd_prep(const float* __restrict__ px,
                                                const float* __restrict__ py,
                                                float* __restrict__ out) {
    int idx = blockIdx.x * TPB + threadIdx.x;       // [0, NB*NPTS)
    if (idx == 0) out[0] = 0.0f;
    float x0 = px[idx * 3 + 0], x1 = px[idx * 3 + 1], x2 = px[idx * 3 + 2];
    g_xs4[idx * 4 + 0] = -2.0f * x0;
    g_xs4[idx * 4 + 1] = -2.0f * x1;
    g_xs4[idx * 4 + 2] = -2.0f * x2;
    g_xs4[idx * 4 + 3] = x0 * x0 + x1 * x1 + x2 * x2;
    g_satl[idx] = 1.0f;                              // max_size/N == 1 (N == M)
    float y0 = py[idx * 3 + 0], y1 = py[idx * 3 + 1], y2 = py[idx * 3 + 2];
    g_ys4[idx * 4 + 0] = y0;
    g_ys4[idx * 4 + 1] = y1;
    g_ys4[idx * 4 + 2] = y2;
    g_ys4[idx * 4 + 3] = 1.0f;
    g_ynorm[idx] = y0 * y0 + y1 * y1 + y2 * y2;
    g_satr[idx] = 1.0f;
    g_lsr[idx]  = 0.0f;                              // log2(1 + 1e-30)
    g_ssum[idx] = 0.0f;
    g_wd2[idx]  = 0.0f;
}

// Pass A: per wave one 16-row block; flash max/sum then column accumulation.
__global__ __launch_bounds__(TPB) void emd_passA(float level2) {
    __shared__ float ys4s[MPTS * 4];   // 32 KB
    __shared__ float yls[MPTS];        // |y|^2*level2 + lsr2
    __shared__ float ynvs[MPTS];
    __shared__ float csum[MPTS];
    __shared__ float cwd2[MPTS];

    const int tid  = threadIdx.x;
    const int b    = blockIdx.x >> 4;          // /WGPB
    const int rg   = blockIdx.x & (WGPB - 1);
    const int lane = tid & 31;
    const int wv   = tid >> 5;
    const int half = lane >> 4;
    const int l16  = lane & 15;
    const int yb   = b * MPTS;

    for (int m = tid; m < MPTS; m += TPB) {
        float4 y = ((const float4*)g_ys4)[yb + m];
        ((float4*)ys4s)[m] = y;
        float ynv = g_ynorm[yb + m];
        ynvs[m] = ynv;
        yls[m]  = fmaf(ynv, level2, g_lsr[yb + m]);
        csum[m] = 0.0f;
        cwd2[m] = 0.0f;
    }
    __syncthreads();

    const int rb = rg * ROWS_PER_WG + wv * 16;
    // A fragment (lanes 0-15: K0,K1; lanes 16-31: K2,K3; M = lane%16)
    v2f afr;
    afr.x = g_xs4[(b * NPTS + rb + l16) * 4 + half * 2 + 0];
    afr.y = g_xs4[(b * NPTS + rb + l16) * 4 + half * 2 + 1];
    const v8f cz = {0.f, 0.f, 0.f, 0.f, 0.f, 0.f, 0.f, 0.f};

    // pass 1: online row max / exp2-sum
    float rm[8], rs[8];
#pragma unroll
    for (int r = 0; r < 8; ++r) { rm[r] = -3.0e38f; rs[r] = 0.0f; }
    for (int ct = 0; ct < CTILES; ++ct) {
        const int col = ct * 16 + l16;
        v2f bfr;
        bfr.x = ys4s[col * 4 + half * 2 + 0];
        bfr.y = ys4s[col * 4 + half * 2 + 1];
        v8f dp = __builtin_amdgcn_wmma_f32_16x16x4_f32(
            false, afr, false, bfr, (short)0, cz, false, false);
        const float yl = yls[col];
#pragma unroll
        for (int r = 0; r < 8; ++r) {
            float lg = fmaf(dp[r], level2, yl);
            float nm = fmaxf(rm[r], lg);
            rs[r] = rs[r] * EXP2(rm[r] - nm) + EXP2(lg - nm);
            rm[r] = nm;
        }
    }
#pragma unroll
    for (int off = 1; off <= 8; off <<= 1) {
#pragma unroll
        for (int r = 0; r < 8; ++r) {
            float om = __shfl_xor(rm[r], off, 32);
            float os = __shfl_xor(rs[r], off, 32);
            float nm = fmaxf(rm[r], om);
            rs[r] = rs[r] * EXP2(rm[r] - nm) + os * EXP2(om - nm);
            rm[r] = nm;
        }
    }
    float winv[8];
#pragma unroll
    for (int r = 0; r < 8; ++r) {
        const int row = b * NPTS + rb + r + half * 8;
        winv[r] = g_satl[row] / rs[r];               // rs >= 1
    }
    if (l16 == 0) {
#pragma unroll
        for (int r = 0; r < 8; ++r) {
            const int row = b * NPTS + rb + r + half * 8;
            g_rowM[row] = rm[r];
            g_winv[row] = winv[r];
        }
    }

    // pass 2: column sums of w and w*d2 (d2 = dp + |y|^2 folded)
    for (int ct = 0; ct < CTILES; ++ct) {
        const int col = ct * 16 + l16;
        v2f bfr;
        bfr.x = ys4s[col * 4 + half * 2 + 0];
        bfr.y = ys4s[col * 4 + half * 2 + 1];
        v8f dp = __builtin_amdgcn_wmma_f32_16x16x4_f32(
            false, afr, false, bfr, (short)0, cz, false, false);
        const float yl = yls[col];
        float cs = 0.0f, cwp = 0.0f;
#pragma unroll
        for (int r = 0; r < 8; ++r) {
            float w = EXP2(fmaf(dp[r], level2, yl) - rm[r]) * winv[r];
            cs  += w;
            cwp  = fmaf(w, dp[r], cwp);
        }
        atomicAdd(&csum[col], cs);                      // ds_add_f32
        atomicAdd(&cwd2[col], fmaf(ynvs[col], cs, cwp));
    }
    __syncthreads();
    for (int m = tid; m < MPTS; m += TPB) {
        atomicAdd(&g_ssum[yb + m], csum[m]);
        atomicAdd(&g_wd2[yb + m], cwd2[m]);
    }
}

// Mid: column rescale + satr update + loss accumulation.
__global__ __launch_bounds__(TPB) void emd_mid(float* __restrict__ out) {
    __shared__ float red[TPB / 32];
    const int idx  = blockIdx.x * TPB + threadIdx.x;   // [0, NB*MPTS)
    const int lane = threadIdx.x & 31;
    const int wv   = threadIdx.x >> 5;
    float cs  = g_ssum[idx];
    float sr  = g_satr[idx];
    float fac = fminf(sr / (cs + 1e-9f), 1.0f);
    g_sfac[idx] = fac;
    g_satr[idx] = fmaxf(sr - cs * fac, 0.0f);
    float part = fac * g_wd2[idx];
#pragma unroll
    for (int off = 16; off > 0; off >>= 1) part += __shfl_xor(part, off, 32);
    if (lane == 0) red[wv] = part;
    __syncthreads();
    if (wv == 0) {
        float v = (lane < TPB / 32) ? red[lane] : 0.0f;
#pragma unroll
        for (int off = 16; off > 0; off >>= 1) v += __shfl_xor(v, off, 32);
        if (lane == 0) atomicAdd(out, v * (1.0f / (float)NB));
    }
}

// Pass C: row sums of w*fac -> satl update.
__global__ __launch_bounds__(TPB) void emd_passC(float level2) {
    __shared__ float ys4s[MPTS * 4];
    __shared__ float yls[MPTS];
    __shared__ float facs[MPTS];

    const int tid  = threadIdx.x;
    const int b    = blockIdx.x >> 4;
    const int rg   = blockIdx.x & (WGPB - 1);
    const int lane = tid & 31;
    const int wv   = tid >> 5;
    const int half = lane >> 4;
    const int l16  = lane & 15;
    const int yb   = b * MPTS;

    for (int m = tid; m < MPTS; m += TPB) {
        ((float4*)ys4s)[m] = ((const float4*)g_ys4)[yb + m];
        yls[m]  = fmaf(g_ynorm[yb + m], level2, g_lsr[yb + m]);
        facs[m] = g_sfac[yb + m];
    }
    __syncthreads();

    const int rb = rg * ROWS_PER_WG + wv * 16;
    v2f afr;
    afr.x = g_xs4[(b * NPTS + rb + l16) * 4 + half * 2 + 0];
    afr.y = g_xs4[(b * NPTS + rb + l16) * 4 + half * 2 + 1];
    const v8f cz = {0.f, 0.f, 0.f, 0.f, 0.f, 0.f, 0.f, 0.f};

    float rm[8], winv[8], rsw[8];
#pragma unroll
    for (int r = 0; r < 8; ++r) {
        const int row = b * NPTS + rb + r + half * 8;
        rm[r]   = g_rowM[row];
        winv[r] = g_winv[row];
        rsw[r]  = 0.0f;
    }
    for (int ct = 0; ct < CTILES; ++ct) {
        const int col = ct * 16 + l16;
        v2f bfr;
        bfr.x = ys4s[col * 4 + half * 2 + 0];
        bfr.y = ys4s[col * 4 + half * 2 + 1];
        v8f dp = __builtin_amdgcn_wmma_f32_16x16x4_f32(
            false, afr, false, bfr, (short)0, cz, false, false);
        const float yl  = yls[col];
        const float fac = facs[col];
#pragma unroll
        for (int r = 0; r < 8; ++r) {
            float w = EXP2(fmaf(dp[r], level2, yl) - rm[r]) * winv[r];
            rsw[r] = fmaf(w, fac, rsw[r]);
        }
    }
#pragma unroll
    for (int off = 1; off <= 8; off <<= 1) {
#pragma unroll
        for (int r = 0; r < 8; ++r) rsw[r] += __shfl_xor(rsw[r], off, 32);
    }
    if (l16 == 0) {
#pragma unroll
        for (int r = 0; r < 8; ++r) {
            const int row = b * NPTS + rb + r + half * 8;
            g_satl[row] = fmaxf(g_satl[row] - rsw[r], 0.0f);
        }
    }
}

// Post: refresh log2(satr) and zero column accumulators for the next level.
__global__ __launch_bounds__(TPB) void emd_post() {
    const int idx = blockIdx.x * TPB + threadIdx.x;
    g_lsr[idx]  = LOG2(g_satr[idx] + 1e-30f);
    g_ssum[idx] = 0.0f;
    g_wd2[idx]  = 0.0f;
}

extern "C" void kernel_launch(void* const* d_in, const int* in_sizes, int n_in,
                              void* d_out, int out_size, void* d_ws, size_t ws_size,
                              hipStream_t stream) {
    (void)in_sizes; (void)n_in; (void)out_size; (void)d_ws; (void)ws_size;
    const float* px = (const float*)d_in[0];
    const float* py = (const float*)d_in[1];
    float* out = (float*)d_out;

    const int elem_blocks = (NB * NPTS) / TPB;       // 128
    const int tile_blocks = NB * WGPB;               // 256

    emd_prep<<<elem_blocks, TPB, 0, stream>>>(px, py, out);
    for (int lev = 0; lev < LEVELS; ++lev) {
        float level  = (lev == LEVELS - 1) ? 0.0f : (8.0f - 0.25f * (float)lev);
        float level2 = level * 1.44269504088896f;    // base-2 logits
        emd_passA<<<tile_blocks, TPB, 0, stream>>>(level2);
        emd_mid<<<elem_blocks, TPB, 0, stream>>>(out);
        emd_passC<<<tile_blocks, TPB, 0, stream>>>(level2);
        emd_post<<<elem_blocks, TPB, 0, stream>>>();
    }
}